// NerfactoRenderer_6038724018408
// MI455X (gfx1250) — compile-verified
//
#include <hip/hip_runtime.h>
#include <cstdint>

// NeRF volume rendering, MI455X (gfx1250, wave32).
// One wave per ray; 6 chunks of 32 samples; multiplicative wave scan for the
// exclusive transmittance cumprod; async global->LDS staging of the density
// row; non-temporal loads/stores for the 805 MB streaming working set.
// Roofline: ~805 MB @ 23.3 TB/s => ~35 us; ~1 GFLOP => purely bandwidth-bound.

#define S_SAMPLES 192
#define CHUNK     32
#define NCHUNK    (S_SAMPLES / CHUNK)
#define WAVES     8                     // 256-thread block = 8 wave32 = 8 rays
#define EPS_W     1e-10f
#define DELTA_INF 1e10f

typedef __attribute__((address_space(1))) int g_int;   // global (AS1)
typedef __attribute__((address_space(3))) int l_int;   // LDS    (AS3)

__device__ __forceinline__ void async_copy_f32(const float* g, float* l) {
#if __has_builtin(__builtin_amdgcn_global_load_async_to_lds_b32)
  __builtin_amdgcn_global_load_async_to_lds_b32(
      (g_int*)(uintptr_t)g, (l_int*)l, /*offset=*/0, /*cpol=*/0);
#else
  unsigned lds_off = (unsigned)(uintptr_t)(l_int*)l;
  asm volatile("global_load_async_to_lds_b32 %0, %1, off"
               :: "v"(lds_off), "v"(g) : "memory");
#endif
}

__device__ __forceinline__ void wait_async0() {
#if __has_builtin(__builtin_amdgcn_s_wait_asynccnt)
  __builtin_amdgcn_s_wait_asynccnt(0);
#else
  asm volatile("s_wait_asynccnt 0" ::: "memory");
#endif
}

// Broadcast lane 31's float value to all lanes via an SGPR (no DS pipe).
__device__ __forceinline__ float bcast31(float v) {
  return __int_as_float(__builtin_amdgcn_readlane(__float_as_int(v), 31));
}

__global__ __launch_bounds__(WAVES * 32)
void nerf_render_kernel(const float* __restrict__ pos,   // [N,S,3]
                        const float* __restrict__ dens,  // [N,S]
                        const float* __restrict__ rgb,   // [N,S,3]
                        const float* __restrict__ bg,    // [3]
                        float* __restrict__ out_rgb,     // [N,3]
                        float* __restrict__ out_acc,     // [N]
                        float* __restrict__ out_w,       // [N,S]
                        int nrays) {
  __shared__ float s_dens[WAVES][S_SAMPLES];

  const int lane = (int)(threadIdx.x & 31u);
  const int wv   = (int)(threadIdx.x >> 5);
  const int ray  = (int)blockIdx.x * WAVES + wv;
  if (ray >= nrays) return;   // whole-wave granularity; N % 8 == 0 in practice

  const size_t base = (size_t)ray * S_SAMPLES;
  const float* prow = pos  + base * 3;
  const float* crow = rgb  + base * 3;
  const float* drow = dens + base;
  float*       wrow = out_w + base;
  float*       sd   = &s_dens[wv][0];

  const int nlane = (lane + 1) & 31;   // rotate-by-1 index for neighbor fetch

  // Stage the full 768 B density row via async DMA (ASYNCcnt-tracked).
#pragma unroll
  for (int c = 0; c < NCHUNK; ++c)
    async_copy_f32(drow + c * CHUNK + lane, sd + c * CHUNK + lane);

  // Chunk-0 positions load while the async copies are in flight.
  float px = __builtin_nontemporal_load(prow + 3 * lane + 0);
  float py = __builtin_nontemporal_load(prow + 3 * lane + 1);
  float pz = __builtin_nontemporal_load(prow + 3 * lane + 2);

  float carry = 1.0f;                       // transmittance carried over chunks
  float aR = 0.f, aG = 0.f, aB = 0.f, aW = 0.f;

  wait_async0();

#pragma unroll
  for (int c = 0; c < NCHUNK; ++c) {
    const int s = c * CHUNK + lane;

    // Next chunk's position (each position is loaded exactly once per ray).
    float nx = 0.f, ny = 0.f, nz = 0.f;
    if (c + 1 < NCHUNK) {
      const float* np = prow + 3 * (s + CHUNK);
      nx = __builtin_nontemporal_load(np + 0);
      ny = __builtin_nontemporal_load(np + 1);
      nz = __builtin_nontemporal_load(np + 2);
      __builtin_prefetch(crow + 3 * (s + CHUNK), 0, 0);   // rgb one chunk ahead
    }

    // pos[s+1]: one rotate-shuffle of a merged register.
    // m = (lane==0) ? next-chunk value : this-chunk value; rotating by 1 gives
    // lane i<31 -> cur[i+1], lane 31 -> next[0]. (c==5/lane31 is overridden.)
    float mx = (lane == 0) ? nx : px;
    float my = (lane == 0) ? ny : py;
    float mz = (lane == 0) ? nz : pz;
    float qx = __shfl(mx, nlane, 32);
    float qy = __shfl(my, nlane, 32);
    float qz = __shfl(mz, nlane, 32);

    float dx = qx - px, dy = qy - py, dz = qz - pz;
    float delta = sqrtf(dx * dx + dy * dy + dz * dz);
    if (c == NCHUNK - 1 && lane == 31) delta = DELTA_INF;  // last sample

    float sigma = fmaxf(sd[s], 0.0f);        // relu(density)
    float t     = __expf(-sigma * delta);    // = 1 - alpha
    float alpha = 1.0f - t;
    float tp    = t + EPS_W;                 // 1 - alpha + eps

    // Inclusive multiplicative Hillis-Steele scan across the wave (5 steps).
    float sc = tp;
#pragma unroll
    for (int off = 1; off < 32; off <<= 1) {
      float u = __shfl_up(sc, off, 32);
      if (lane >= off) sc *= u;
    }
    float ex = __shfl_up(sc, 1, 32);         // exclusive scan
    if (lane == 0) ex = 1.0f;

    float T = carry * ex;                    // transmittance
    float w = alpha * T;                     // weight
    __builtin_nontemporal_store(w, wrow + s);

    float cr = __builtin_nontemporal_load(crow + 3 * s + 0);
    float cg = __builtin_nontemporal_load(crow + 3 * s + 1);
    float cb = __builtin_nontemporal_load(crow + 3 * s + 2);
    aR += w * cr; aG += w * cg; aB += w * cb; aW += w;

    carry *= bcast31(sc);                    // SGPR broadcast, no DS pipe
    px = nx; py = ny; pz = nz;
  }

  // Wave-wide sum reduction (5 shuffle steps per channel).
#pragma unroll
  for (int off = 16; off > 0; off >>= 1) {
    aR += __shfl_down(aR, off, 32);
    aG += __shfl_down(aG, off, 32);
    aB += __shfl_down(aB, off, 32);
    aW += __shfl_down(aW, off, 32);
  }

  if (lane == 0) {
    float inv = 1.0f - aW;
    out_rgb[3 * (size_t)ray + 0] = aR + bg[0] * inv;
    out_rgb[3 * (size_t)ray + 1] = aG + bg[1] * inv;
    out_rgb[3 * (size_t)ray + 2] = aB + bg[2] * inv;
    out_acc[ray] = aW;
  }
}

extern "C" void kernel_launch(void* const* d_in, const int* in_sizes, int n_in,
                              void* d_out, int out_size, void* d_ws, size_t ws_size,
                              hipStream_t stream) {
  const float* pos  = (const float*)d_in[0];   // [N,S,3]
  const float* dens = (const float*)d_in[1];   // [N,S,1]
  const float* rgb  = (const float*)d_in[2];   // [N,S,3]
  const float* bg   = (const float*)d_in[3];   // [3]

  const int N = in_sizes[1] / S_SAMPLES;

  float* out_rgb = (float*)d_out;              // [N,3]
  float* out_acc = out_rgb + (size_t)N * 3;    // [N]
  float* out_w   = out_acc + N;                // [N,S]

  dim3 block(WAVES * 32);
  dim3 grid((N + WAVES - 1) / WAVES);
  hipLaunchKernelGGL(nerf_render_kernel, grid, block, 0, stream,
                     pos, dens, rgb, bg, out_rgb, out_acc, out_w, N);
}